// GAT_edge_41137196761625
// MI455X (gfx1250) — compile-verified
//
#include <hip/hip_runtime.h>
#include <math.h>

// ---------------------------------------------------------------------------
// GAT_edge on gfx1250 (MI455X, wave32).
// Key algebraic optimizations (exact w.r.t. the reference):
//   * a_edge = ea @ We reduced against att_edge  ==>  ea @ V, V = fold(We,att_edge) (64x8)
//   * self-loop edge_attr mean never materialized: mean@V = segsum(a_edge)/cnt
//   * bias cancels inside training-mode BatchNorm (mu shifts by bias)
// Node projection h@W uses native fp32 WMMA: v_wmma_f32_16x16x4_f32.
// ---------------------------------------------------------------------------

#define N_NODES 10000
#define E_EDGES 200000
#define DIN     64
#define HID     128
#define NHEAD   8
#define CCH     16
#define NGRAPH  32
#define BN_EPS  1e-5f
#define SLOPE   0.2f

typedef float v2f __attribute__((ext_vector_type(2)));
typedef float v8f __attribute__((ext_vector_type(8)));

__device__ __forceinline__ void atomicMaxFloat(float* addr, float v) {
  if (v >= 0.0f) atomicMax((int*)addr, __float_as_int(v));
  else           atomicMin((unsigned int*)addr, __float_as_uint(v));
}

// -------------------------- small utility kernels --------------------------
__global__ void fill_f32(float* p, float v, int n) {
  int i = blockIdx.x * blockDim.x + threadIdx.x;
  if (i < n) p[i] = v;
}

__global__ void deg_kernel(const int* __restrict__ dst, float* __restrict__ cnt, int e) {
  int i = blockIdx.x * blockDim.x + threadIdx.x;
  if (i < e) atomicAdd(&cnt[dst[i]], 1.0f);
}

// V[d][h] = sum_c We[d, h*16+c] * att_edge[h, c]
__global__ void fold_edge_kernel(const float* __restrict__ We,
                                 const float* __restrict__ att_edge,
                                 float* __restrict__ V) {
  int i = blockIdx.x * blockDim.x + threadIdx.x;
  if (i >= DIN * NHEAD) return;
  int d = i / NHEAD, h = i % NHEAD;
  float s = 0.0f;
  #pragma unroll
  for (int c = 0; c < CCH; ++c) s += We[d * HID + h * CCH + c] * att_edge[h * CCH + c];
  V[d * NHEAD + h] = s;
}

// aE[e,h] = edge_attr[e,:] @ V[:,h]; also aEsum[dst] += aE (self-loop numerator)
__global__ void edge_logit_kernel(const float* __restrict__ ea,
                                  const int* __restrict__ dst,
                                  const float* __restrict__ V,
                                  float* __restrict__ aE,
                                  float* __restrict__ aEsum, int e) {
  __shared__ float sV[DIN * NHEAD];
  for (int i = threadIdx.x; i < DIN * NHEAD; i += blockDim.x) sV[i] = V[i];
  __syncthreads();
  int idx = blockIdx.x * blockDim.x + threadIdx.x;
  if (idx >= e) return;
  const float* row = ea + (size_t)idx * DIN;
  float acc[NHEAD];
  #pragma unroll
  for (int h = 0; h < NHEAD; ++h) acc[h] = 0.0f;
  for (int d = 0; d < DIN; ++d) {
    float x = row[d];
    #pragma unroll
    for (int h = 0; h < NHEAD; ++h) acc[h] = fmaf(x, sV[d * NHEAD + h], acc[h]);
  }
  int dd = dst[idx];
  #pragma unroll
  for (int h = 0; h < NHEAD; ++h) {
    aE[idx * NHEAD + h] = acc[h];
    atomicAdd(&aEsum[dd * NHEAD + h], acc[h]);
  }
}

// ---------------- node projection GEMM: hh[n,128] = h[n,din] @ W[din,128] ---
// One wave computes a 16x16 output tile via v_wmma_f32_16x16x4_f32.
// Block = 256 threads = 8 waves -> 16 rows x 128 cols per block.
// f32 A frag (16x4): lanes 0-15 hold {A[l,k],A[l,k+1]}, lanes 16-31 {A[l,k+2],A[l,k+3]}
// f32 B frag (4x16): lanes 0-15 hold {B[k,l],B[k+1,l]},  lanes 16-31 {B[k+2,l],B[k+3,l]}
// D: vgpr j of lane l holds D[8*(l>=16)+j][l&15]  (ISA 7.12.2)
__global__ void gemm_wmma_kernel(const float* __restrict__ A,
                                 const float* __restrict__ W,
                                 float* __restrict__ D, int n, int din) {
  int row0 = blockIdx.x * 16;
  int wave = threadIdx.x >> 5;   // 0..7 -> column tile
  int lane = threadIdx.x & 31;
  int half = lane >> 4;          // 0/1
  int l    = lane & 15;
  int col0 = wave * 16;

  int r = row0 + l;
  if (r >= n) r = n - 1;         // clamp: duplicated rows never stored
  const float* arow = A + (size_t)r * din;

  v8f acc = {};
  for (int k = 0; k < din; k += 4) {
    v2f a;
    a.x = arow[k + 2 * half];
    a.y = arow[k + 2 * half + 1];
    const float* bp = W + (size_t)(k + 2 * half) * HID + col0 + l;
    v2f b;
    b.x = bp[0];
    b.y = bp[HID];
    acc = __builtin_amdgcn_wmma_f32_16x16x4_f32(false, a, false, b,
                                                (short)0, acc, false, false);
  }

  int col = col0 + l;
  int rbase = row0 + 8 * half;
  #pragma unroll
  for (int j = 0; j < 8; ++j) {
    int rr = rbase + j;
    if (rr < n) D[(size_t)rr * HID + col] = acc[j];
  }
}

// a_src[n,h], a_dst[n,h] from hh
__global__ void node_att_kernel(const float* __restrict__ hh,
                                const float* __restrict__ att_src,
                                const float* __restrict__ att_dst,
                                float* __restrict__ a_src,
                                float* __restrict__ a_dst) {
  int i = blockIdx.x * blockDim.x + threadIdx.x;
  if (i >= N_NODES * NHEAD) return;
  int nn = i / NHEAD, h = i % NHEAD;
  const float* p = hh + (size_t)nn * HID + h * CCH;
  float s = 0.0f, d = 0.0f;
  #pragma unroll
  for (int c = 0; c < CCH; ++c) {
    s = fmaf(p[c], att_src[h * CCH + c], s);
    d = fmaf(p[c], att_dst[h * CCH + c], d);
  }
  a_src[i] = s;
  a_dst[i] = d;
}

// pass 1: logits (leaky_relu) + per-dst max via atomic bit-trick
__global__ void softmax_max_kernel(const int* __restrict__ src,
                                   const int* __restrict__ dst,
                                   const float* __restrict__ aE,
                                   const float* __restrict__ aEsum,
                                   const float* __restrict__ cnt,
                                   const float* __restrict__ a_src,
                                   const float* __restrict__ a_dst,
                                   float* __restrict__ wbuf,
                                   float* __restrict__ m) {
  int i = blockIdx.x * blockDim.x + threadIdx.x;
  const int total = (E_EDGES + N_NODES) * NHEAD;
  if (i >= total) return;
  int e = i / NHEAD, h = i % NHEAD;
  int s, d;
  float ae;
  if (e < E_EDGES) {
    s = src[e]; d = dst[e];
    ae = aE[(size_t)e * NHEAD + h];
  } else {
    int nn = e - E_EDGES;
    s = nn; d = nn;
    ae = aEsum[nn * NHEAD + h] / fmaxf(cnt[nn], 1.0f);
  }
  float lg = a_src[s * NHEAD + h] + a_dst[d * NHEAD + h] + ae;
  lg = (lg > 0.0f) ? lg : SLOPE * lg;
  wbuf[i] = lg;
  atomicMaxFloat(&m[d * NHEAD + h], lg);
}

// pass 2: w = exp(logit - m[dst]); denom[dst] += w
__global__ void softmax_exp_kernel(const int* __restrict__ dst,
                                   const float* __restrict__ m,
                                   float* __restrict__ wbuf,
                                   float* __restrict__ denom) {
  int i = blockIdx.x * blockDim.x + threadIdx.x;
  const int total = (E_EDGES + N_NODES) * NHEAD;
  if (i >= total) return;
  int e = i / NHEAD, h = i % NHEAD;
  int d = (e < E_EDGES) ? dst[e] : (e - E_EDGES);
  float w = __expf(wbuf[i] - m[d * NHEAD + h]);
  wbuf[i] = w;
  atomicAdd(&denom[d * NHEAD + h], w);
}

// pass 3: out[dst] += alpha * hh[src]; head-mean folded in for the last layer
__global__ void aggregate_kernel(const int* __restrict__ src,
                                 const int* __restrict__ dst,
                                 const float* __restrict__ wbuf,
                                 const float* __restrict__ denom,
                                 const float* __restrict__ hh,
                                 float* __restrict__ out, int concat) {
  int i = blockIdx.x * blockDim.x + threadIdx.x;
  const int total = (E_EDGES + N_NODES) * NHEAD;
  if (i >= total) return;
  int e = i / NHEAD, h = i % NHEAD;
  int s = (e < E_EDGES) ? src[e] : (e - E_EDGES);
  int d = (e < E_EDGES) ? dst[e] : (e - E_EDGES);
  float alpha = wbuf[i] / denom[d * NHEAD + h];
  const float* hp = hh + (size_t)s * HID + h * CCH;
  if (concat) {
    float* op = out + (size_t)d * HID + h * CCH;
    #pragma unroll
    for (int c = 0; c < CCH; ++c) atomicAdd(&op[c], alpha * hp[c]);
  } else {
    float a = alpha * (1.0f / NHEAD);
    float* op = out + (size_t)d * CCH;
    #pragma unroll
    for (int c = 0; c < CCH; ++c) atomicAdd(&op[c], a * hp[c]);
  }
}

// BN batch stats: one block per channel, no atomics
__global__ void bn_stats_kernel(const float* __restrict__ out,
                                float* __restrict__ stats, int odim) {
  int ch = blockIdx.x;
  __shared__ float ss[256];
  __shared__ float sq[256];
  float s = 0.0f, q = 0.0f;
  for (int nn = threadIdx.x; nn < N_NODES; nn += blockDim.x) {
    float v = out[(size_t)nn * odim + ch];
    s += v; q = fmaf(v, v, q);
  }
  ss[threadIdx.x] = s; sq[threadIdx.x] = q;
  __syncthreads();
  for (int st = 128; st > 0; st >>= 1) {
    if (threadIdx.x < st) {
      ss[threadIdx.x] += ss[threadIdx.x + st];
      sq[threadIdx.x] += sq[threadIdx.x + st];
    }
    __syncthreads();
  }
  if (threadIdx.x == 0) { stats[ch] = ss[0]; stats[odim + ch] = sq[0]; }
}

// normalize (+gamma,beta) then ELU  (bias cancels inside batch-stat BN)
__global__ void bn_elu_kernel(const float* __restrict__ out,
                              const float* __restrict__ stats,
                              const float* __restrict__ gamma,
                              const float* __restrict__ beta,
                              float* __restrict__ hn, int odim) {
  int i = blockIdx.x * blockDim.x + threadIdx.x;
  if (i >= N_NODES * odim) return;
  int ch = i % odim;
  float mu  = stats[ch] * (1.0f / N_NODES);
  float var = stats[odim + ch] * (1.0f / N_NODES) - mu * mu;
  float y = (out[i] - mu) * rsqrtf(var + BN_EPS) * gamma[ch] + beta[ch];
  hn[i] = (y > 0.0f) ? y : (__expf(y) - 1.0f);
}

__global__ void pool_kernel(const float* __restrict__ h, const int* __restrict__ batch,
                            float* __restrict__ gsum, float* __restrict__ gcnt) {
  int nn = blockIdx.x * blockDim.x + threadIdx.x;
  if (nn >= N_NODES) return;
  int b = batch[nn];
  atomicAdd(&gcnt[b], 1.0f);
  #pragma unroll
  for (int c = 0; c < CCH; ++c)
    atomicAdd(&gsum[b * CCH + c], h[(size_t)nn * CCH + c]);
}

__global__ void pool_div_kernel(const float* __restrict__ gsum,
                                const float* __restrict__ gcnt,
                                float* __restrict__ out) {
  int i = blockIdx.x * blockDim.x + threadIdx.x;
  if (i >= NGRAPH * CCH) return;
  out[i] = gsum[i] / fmaxf(gcnt[i / CCH], 1.0f);
}

// ---------------------------------------------------------------------------
extern "C" void kernel_launch(void* const* d_in, const int* in_sizes, int n_in,
                              void* d_out, int out_size, void* d_ws, size_t ws_size,
                              hipStream_t stream) {
  (void)in_sizes; (void)n_in; (void)out_size; (void)ws_size;

  const float* x     = (const float*)d_in[0];
  const int*   src   = (const int*)d_in[1];
  const int*   dst   = src + E_EDGES;
  const float* eattr = (const float*)d_in[2];
  const int*   batch = (const int*)d_in[3];

  // Workspace carve-up (floats)
  float* ws = (float*)d_ws;
  size_t off = 0;
  float* cnt   = ws + off; off += N_NODES;
  float* aEsum = ws + off; off += (size_t)N_NODES * NHEAD;
  float* aE    = ws + off; off += (size_t)E_EDGES * NHEAD;
  float* Vbuf  = ws + off; off += DIN * NHEAD;
  float* hh    = ws + off; off += (size_t)N_NODES * HID;
  float* asrcb = ws + off; off += (size_t)N_NODES * NHEAD;
  float* adstb = ws + off; off += (size_t)N_NODES * NHEAD;
  float* mbuf  = ws + off; off += (size_t)N_NODES * NHEAD;
  float* denom = ws + off; off += (size_t)N_NODES * NHEAD;
  float* wbuf  = ws + off; off += (size_t)(E_EDGES + N_NODES) * NHEAD;
  float* outb  = ws + off; off += (size_t)N_NODES * HID;
  float* hA    = ws + off; off += (size_t)N_NODES * HID;
  float* hB    = ws + off; off += (size_t)N_NODES * HID;
  float* stats = ws + off; off += 2 * HID;
  float* gsum  = ws + off; off += NGRAPH * CCH;
  float* gcnt  = ws + off; off += NGRAPH;

  const int TOT = (E_EDGES + N_NODES) * NHEAD;

  hipMemsetAsync(cnt, 0, N_NODES * sizeof(float), stream);
  deg_kernel<<<(E_EDGES + 255) / 256, 256, 0, stream>>>(dst, cnt, E_EDGES);

  const float* hcur = x;
  int din = DIN;
  for (int l = 0; l < 4; ++l) {
    const float* W     = (const float*)d_in[4 + 8 * l + 0];
    const float* We    = (const float*)d_in[4 + 8 * l + 1];
    const float* attd  = (const float*)d_in[4 + 8 * l + 2];
    const float* atte  = (const float*)d_in[4 + 8 * l + 3];
    const float* atts  = (const float*)d_in[4 + 8 * l + 4];
    const float* beta  = (const float*)d_in[4 + 8 * l + 5];
    /* bias (idx 6) cancels in batch-stat BatchNorm */
    const float* gamma = (const float*)d_in[4 + 8 * l + 7];
    int concat = (l < 3);
    int odim = concat ? HID : CCH;

    hipMemsetAsync(aEsum, 0, (size_t)N_NODES * NHEAD * sizeof(float), stream);
    hipMemsetAsync(denom, 0, (size_t)N_NODES * NHEAD * sizeof(float), stream);
    hipMemsetAsync(outb, 0, (size_t)N_NODES * odim * sizeof(float), stream);
    fill_f32<<<(N_NODES * NHEAD + 255) / 256, 256, 0, stream>>>(mbuf, -3.0e38f, N_NODES * NHEAD);

    fold_edge_kernel<<<2, 256, 0, stream>>>(We, atte, Vbuf);
    edge_logit_kernel<<<(E_EDGES + 255) / 256, 256, 0, stream>>>(eattr, dst, Vbuf, aE, aEsum, E_EDGES);
    gemm_wmma_kernel<<<(N_NODES + 15) / 16, 256, 0, stream>>>(hcur, W, hh, N_NODES, din);
    node_att_kernel<<<(N_NODES * NHEAD + 255) / 256, 256, 0, stream>>>(hh, atts, attd, asrcb, adstb);
    softmax_max_kernel<<<(TOT + 255) / 256, 256, 0, stream>>>(src, dst, aE, aEsum, cnt,
                                                              asrcb, adstb, wbuf, mbuf);
    softmax_exp_kernel<<<(TOT + 255) / 256, 256, 0, stream>>>(dst, mbuf, wbuf, denom);
    aggregate_kernel<<<(TOT + 255) / 256, 256, 0, stream>>>(src, dst, wbuf, denom, hh, outb, concat);
    bn_stats_kernel<<<odim, 256, 0, stream>>>(outb, stats, odim);
    float* hnext = (l & 1) ? hB : hA;
    bn_elu_kernel<<<((N_NODES * odim) + 255) / 256, 256, 0, stream>>>(outb, stats, gamma, beta,
                                                                     hnext, odim);
    hcur = hnext;
    din = HID;
  }

  hipMemsetAsync(gsum, 0, (NGRAPH * CCH + NGRAPH) * sizeof(float), stream);
  pool_kernel<<<(N_NODES + 255) / 256, 256, 0, stream>>>(hcur, batch, gsum, gcnt);
  pool_div_kernel<<<(NGRAPH * CCH + 255) / 256, 256, 0, stream>>>(gsum, gcnt, (float*)d_out);
}